// GNNStack_30769145708974
// MI455X (gfx1250) — compile-verified
//
#include <hip/hip_runtime.h>
#include <hip/hip_bf16.h>

typedef __attribute__((ext_vector_type(2))) float v2f;
typedef __attribute__((ext_vector_type(8))) float v8f;

#define NEG_GAT 0.2f
#define NEG_ACT 0.01f

__device__ __forceinline__ float lrelu(float x, float sl) { return x > 0.f ? x : sl * x; }

// Native f32 atomic max/add at device scope (no CAS loop) — CDNA5 GLOBAL atomics.
__device__ __forceinline__ void atomic_fmax_g(float* p, float v) {
    asm volatile("global_atomic_max_num_f32 %0, %1, off scope:SCOPE_DEV"
                 :: "v"(p), "v"(v) : "memory");
}
__device__ __forceinline__ void atomic_fadd_g(float* p, float v) {
    asm volatile("global_atomic_add_f32 %0, %1, off scope:SCOPE_DEV"
                 :: "v"(p), "v"(v) : "memory");
}

// H[Nrows x M] = X[Nrows x K] @ W[M x K]^T + bias, f32 WMMA 16x16x4.
// One 16x16 output tile per wave. Nrows % 16 == 0, M % 16 == 0, K % 4 == 0.
__global__ void gemm_bias_wmma(const float* __restrict__ X, const float* __restrict__ W,
                               const float* __restrict__ bias, float* __restrict__ H,
                               int Nrows, int K, int M) {
    const int lane = threadIdx.x & 31;
    const int wid  = threadIdx.x >> 5;
    const int wavesPerBlock = blockDim.x >> 5;
    const int colTiles = M >> 4;
    const int totalTiles = (Nrows >> 4) * colTiles;
    const int tile = blockIdx.x * wavesPerBlock + wid;   // uniform per wave
    if (tile >= totalTiles) return;                      // whole-wave exit, EXEC stays all-1s
    const int rt = tile / colTiles;
    const int ct = tile - rt * colTiles;

    const int l15  = lane & 15;
    const int khalf = lane >> 4;                 // 0 (lanes 0-15) or 1 (lanes 16-31)
    const int kb   = khalf << 1;                 // K offset 0 or 2 within each 16x16x4 step
    const int mrow = (rt << 4) + l15;            // A: M = lane&15
    const int ncol = (ct << 4) + l15;            // B: N = lane&15
    const float* __restrict__ xrow = X + (size_t)mrow * K + kb;
    const float* __restrict__ wrow = W + (size_t)ncol * K + kb;

    v8f c = {};
    #pragma unroll 4
    for (int k0 = 0; k0 < K; k0 += 4) {
        v2f a = *(const v2f*)(xrow + k0);        // {X[m,k0+kb], X[m,k0+kb+1]}
        v2f b = *(const v2f*)(wrow + k0);        // {W[n,k0+kb], W[n,k0+kb+1]}
        c = __builtin_amdgcn_wmma_f32_16x16x4_f32(false, a, false, b, (short)0, c,
                                                  false, false);
    }

    // C/D layout: VGPR j -> (M = j + 8*khalf, N = lane&15)
    const float bj = bias[ncol];
    const int rbase = (rt << 4) + (khalf << 3);
    #pragma unroll
    for (int j = 0; j < 8; ++j) {
        H[(size_t)(rbase + j) * M + ncol] = c[j] + bj;
    }
}

__global__ void init_mza(float* __restrict__ m, float* __restrict__ z,
                         float* __restrict__ acc, int n) {
    int i = blockIdx.x * blockDim.x + threadIdx.x;
    if (i < n) { m[i] = -3.402823466e38f; z[i] = 0.f; acc[i] = 0.f; }
}

// Pass 1: per-(edge,channel) attention logit -> atomic max into m[dst,c].
// One wave per edge, 4 channels per lane (128 channels).
__global__ void edge_max_k(const float* __restrict__ h, const int* __restrict__ src,
                           const int* __restrict__ dst, const float* __restrict__ attl,
                           const float* __restrict__ attr, float* __restrict__ m, int E) {
    const int lane = threadIdx.x & 31;
    const int wid  = threadIdx.x >> 5;
    const int e = blockIdx.x * (blockDim.x >> 5) + wid;
    if (e >= E) return;
    const int s = src[e], d = dst[e];
    const int c4 = lane << 2;
    float4 hs = *(const float4*)&h[(size_t)s * 128 + c4];
    float4 hd = *(const float4*)&h[(size_t)d * 128 + c4];
    float4 al = *(const float4*)&attl[c4];
    float4 ar = *(const float4*)&attr[c4];
    float* mp = m + (size_t)d * 128 + c4;
    atomic_fmax_g(mp + 0, lrelu(al.x * hs.x + ar.x * hd.x, NEG_GAT));
    atomic_fmax_g(mp + 1, lrelu(al.y * hs.y + ar.y * hd.y, NEG_GAT));
    atomic_fmax_g(mp + 2, lrelu(al.z * hs.z + ar.z * hd.z, NEG_GAT));
    atomic_fmax_g(mp + 3, lrelu(al.w * hs.w + ar.w * hd.w, NEG_GAT));
}

// Pass 2: e = exp(s - m[dst]); z[dst,c] += e; acc[dst,c] += e * h[src,c].
// Division by z deferred to finalize (z constant per output slot).
__global__ void edge_accum_k(const float* __restrict__ h, const int* __restrict__ src,
                             const int* __restrict__ dst, const float* __restrict__ attl,
                             const float* __restrict__ attr, const float* __restrict__ m,
                             float* __restrict__ z, float* __restrict__ acc, int E) {
    const int lane = threadIdx.x & 31;
    const int wid  = threadIdx.x >> 5;
    const int e = blockIdx.x * (blockDim.x >> 5) + wid;
    if (e >= E) return;
    const int s = src[e], d = dst[e];
    const int c4 = lane << 2;
    float4 hs = *(const float4*)&h[(size_t)s * 128 + c4];
    float4 hd = *(const float4*)&h[(size_t)d * 128 + c4];
    float4 al = *(const float4*)&attl[c4];
    float4 ar = *(const float4*)&attr[c4];
    float4 mm = *(const float4*)&m[(size_t)d * 128 + c4];
    float e0 = __expf(lrelu(al.x * hs.x + ar.x * hd.x, NEG_GAT) - mm.x);
    float e1 = __expf(lrelu(al.y * hs.y + ar.y * hd.y, NEG_GAT) - mm.y);
    float e2 = __expf(lrelu(al.z * hs.z + ar.z * hd.z, NEG_GAT) - mm.z);
    float e3 = __expf(lrelu(al.w * hs.w + ar.w * hd.w, NEG_GAT) - mm.w);
    float* zp = z   + (size_t)d * 128 + c4;
    float* ap = acc + (size_t)d * 128 + c4;
    atomic_fadd_g(zp + 0, e0);  atomic_fadd_g(ap + 0, e0 * hs.x);
    atomic_fadd_g(zp + 1, e1);  atomic_fadd_g(ap + 1, e1 * hs.y);
    atomic_fadd_g(zp + 2, e2);  atomic_fadd_g(ap + 2, e2 * hs.z);
    atomic_fadd_g(zp + 3, e3);  atomic_fadd_g(ap + 3, e3 * hs.w);
}

// x = leaky_relu(acc / z, slope); isolated nodes (z == 0) -> 0 (empty segment_sum).
__global__ void finalize_k(const float* __restrict__ acc, const float* __restrict__ z,
                           float* __restrict__ x, int n, float slope) {
    int i = blockIdx.x * blockDim.x + threadIdx.x;
    if (i < n) {
        float zz = z[i];
        float v = (zz > 0.f) ? (acc[i] / zz) : 0.f;
        x[i] = lrelu(v, slope);
    }
}

// sig[e] = sigmoid(dot64(xf[src], xf[dst])); one wave per edge, 2 channels/lane.
__global__ void edge_dot_k(const float* __restrict__ xf, const int* __restrict__ src,
                           const int* __restrict__ dst, float* __restrict__ sig, int E) {
    const int lane = threadIdx.x & 31;
    const int wid  = threadIdx.x >> 5;
    const int e = blockIdx.x * (blockDim.x >> 5) + wid;
    if (e >= E) return;
    const int s = src[e], d = dst[e];
    float2 u = *(const float2*)&xf[(size_t)s * 64 + lane * 2];
    float2 v = *(const float2*)&xf[(size_t)d * 64 + lane * 2];
    float accv = u.x * v.x + u.y * v.y;
    #pragma unroll
    for (int off = 16; off; off >>= 1) accv += __shfl_xor(accv, off, 32);
    if (lane == 0) sig[e] = 1.f / (1.f + __expf(-accv));
}

extern "C" void kernel_launch(void* const* d_in, const int* in_sizes, int n_in,
                              void* d_out, int out_size, void* d_ws, size_t ws_size,
                              hipStream_t stream) {
    const float* emb   = (const float*)d_in[0];
    const int*   ei    = (const int*)  d_in[1];
    const float* W0    = (const float*)d_in[2];
    const float* b0    = (const float*)d_in[3];
    const float* attl0 = (const float*)d_in[4];
    const float* attr0 = (const float*)d_in[5];
    const float* W1    = (const float*)d_in[6];
    const float* b1    = (const float*)d_in[7];
    const float* attl1 = (const float*)d_in[8];
    const float* attr1 = (const float*)d_in[9];
    const float* pW1   = (const float*)d_in[10];
    const float* pb1   = (const float*)d_in[11];
    const float* pW2   = (const float*)d_in[12];
    const float* pb2   = (const float*)d_in[13];

    const int N = in_sizes[0] / 128;          // 50000
    const int E = in_sizes[1] / 2;            // 800000
    const int* src = ei;
    const int* dst = ei + E;

    const size_t NF = (size_t)N * 128;
    float* ws  = (float*)d_ws;
    float* h   = ws;                          // [N,128] hidden features
    float* xb  = ws + NF;                     // [N,128] activated features
    float* m   = ws + 2 * NF;                 // [N,128] segment max (reused as p1 [N,64])
    float* z   = ws + 3 * NF;                 // [N,128] segment exp-sum
    float* acc = ws + 4 * NF;                 // [N,128] weighted accumulator
    float* sig = (float*)d_out;               // [E]
    float* xo  = (float*)d_out + E;           // [N,64] final node embeddings

    const int nf = (int)NF;
    const int GEMM_BLK = 128;                 // 4 waves/block, 1 tile/wave
    const int EDGE_BLK = 256;                 // 8 waves/block, 1 edge/wave
    const int edgeBlocks = (E + 7) / 8;
    const int elemBlocks = (nf + 255) / 256;

    auto gemmBlocks = [](int rows, int M) {
        int tiles = (rows >> 4) * (M >> 4);
        return (tiles + 3) / 4;
    };

    // ---- GAT layer 0 ----
    gemm_bias_wmma<<<gemmBlocks(N, 128), GEMM_BLK, 0, stream>>>(emb, W0, b0, h, N, 128, 128);
    init_mza<<<elemBlocks, 256, 0, stream>>>(m, z, acc, nf);
    edge_max_k<<<edgeBlocks, EDGE_BLK, 0, stream>>>(h, src, dst, attl0, attr0, m, E);
    edge_accum_k<<<edgeBlocks, EDGE_BLK, 0, stream>>>(h, src, dst, attl0, attr0, m, z, acc, E);
    finalize_k<<<elemBlocks, 256, 0, stream>>>(acc, z, xb, nf, NEG_ACT);

    // ---- GAT layer 1 ----
    gemm_bias_wmma<<<gemmBlocks(N, 128), GEMM_BLK, 0, stream>>>(xb, W1, b1, h, N, 128, 128);
    init_mza<<<elemBlocks, 256, 0, stream>>>(m, z, acc, nf);
    edge_max_k<<<edgeBlocks, EDGE_BLK, 0, stream>>>(h, src, dst, attl1, attr1, m, E);
    edge_accum_k<<<edgeBlocks, EDGE_BLK, 0, stream>>>(h, src, dst, attl1, attr1, m, z, acc, E);
    finalize_k<<<elemBlocks, 256, 0, stream>>>(acc, z, xb, nf, NEG_ACT);

    // ---- post_mp: Linear(128->64) then Linear(64->64) ----
    gemm_bias_wmma<<<gemmBlocks(N, 64), GEMM_BLK, 0, stream>>>(xb, pW1, pb1, m, N, 128, 64);
    gemm_bias_wmma<<<gemmBlocks(N, 64), GEMM_BLK, 0, stream>>>(m, pW2, pb2, xo, N, 64, 64);

    // ---- edge link scores ----
    edge_dot_k<<<edgeBlocks, EDGE_BLK, 0, stream>>>(xo, src, dst, sig, E);
}